// AdditiveAttentionPooling_24919400251608
// MI455X (gfx1250) — compile-verified
//
#include <hip/hip_runtime.h>
#include <cstdint>
#include <cstddef>

// ---------------- problem constants (derived/validated at launch) ----------
#define TCH    32          // T-rows per chunk staged in LDS (32*1024*4B = 128KB)
#define DMODEL 1024        // feature dim
#define NEGV  -1e9f        // matches reference NEG
#define MAXC   64          // max chunks per batch row (T=2048 -> 64)

typedef __attribute__((ext_vector_type(2))) float v2f;
typedef __attribute__((ext_vector_type(8))) float v8f;
typedef __attribute__((ext_vector_type(4))) int   v4i;

#define GLOBAL_AS __attribute__((address_space(1)))
#define LDS_AS    __attribute__((address_space(3)))

#if __has_builtin(__builtin_amdgcn_wmma_f32_16x16x4_f32)
#define HAVE_WMMA4 1
#endif

#if __has_builtin(__builtin_amdgcn_global_load_async_to_lds_b128)
#define HAVE_ASYNC 1
#endif

#ifdef HAVE_ASYNC
__device__ __forceinline__ void wait_async0() {
#if __has_builtin(__builtin_amdgcn_s_wait_asynccnt)
  __builtin_amdgcn_s_wait_asynccnt(0);
#else
  asm volatile("s_wait_asynccnt 0" ::: "memory");
#endif
}
#endif

// ---------------------------------------------------------------------------
// Kernel 1: one workgroup per (batch b, chunk c of 32 T-rows).
//  Phase 0: async-stage x[b, t0:t0+32, :] (128 KB) into LDS  (x read ONCE)
//  Phase 1: 32 chunk scores  s_t = <x_t, w> + bias  (masked -> NEGV)
//  Phase 2: chunk softmax stats m_c, l_c and unnormalized alphas
//  Phase 3: WMMA f32 16x16x4 weighted accumulation -> partial[1024] to ws
// ---------------------------------------------------------------------------
__global__ __launch_bounds__(256) void app_partial_kernel(
    const float* __restrict__ x, const unsigned char* __restrict__ mask,
    const float* __restrict__ w, const float* __restrict__ bias,
    float* __restrict__ wsAcc, float* __restrict__ wsM, float* __restrict__ wsL,
    int T, int C)
{
  __shared__ float xs[TCH * DMODEL];   // 128 KB chunk of x
  __shared__ float score_s[TCH];
  __shared__ float alpha_s[TCH];

  const int tid  = threadIdx.x;
  const int lane = tid & 31;
  const int wv   = tid >> 5;           // 8 waves per workgroup (wave32)
  const int b    = blockIdx.y;
  const int c    = blockIdx.x;
  const int t0   = c * TCH;
  const float* xc = x + ((size_t)b * T + t0) * DMODEL;

  // ---- Phase 0: stage chunk into LDS -------------------------------------
#ifdef HAVE_ASYNC
  #pragma unroll
  for (int i = 0; i < 32; ++i) {
    const int off = (tid + 256 * i) * 4;   // float index, 16B granules, coalesced sweeps
    __builtin_amdgcn_global_load_async_to_lds_b128(
        (GLOBAL_AS v4i*)(xc + off),
        (LDS_AS v4i*)(xs + off), 0, 0);
  }
  wait_async0();
#else
  #pragma unroll
  for (int i = 0; i < 32; ++i) {
    const int off = (tid + 256 * i) * 4;
    *(float4*)(xs + off) = *(const float4*)(xc + off);
  }
#endif
  __syncthreads();

  // ---- Phase 1: scores. Wave wv computes rows t = 4*wv .. 4*wv+3 ----------
  // lane l owns d in [32*l, 32*l+32): contiguous 128B per lane.
  float4 wr[8];
  {
    const float4* wv4 = (const float4*)(w + 32 * lane);
    #pragma unroll
    for (int j = 0; j < 8; ++j) wr[j] = wv4[j];
  }
  const float bb = bias[0];

  #pragma unroll
  for (int r = 0; r < 4; ++r) {
    const int t = wv * 4 + r;
    const float4* xr4 = (const float4*)(xs + t * DMODEL + 32 * lane);
    float s = 0.f;
    #pragma unroll
    for (int j = 0; j < 8; ++j) {
      const float4 xv = xr4[j];
      s += xv.x * wr[j].x + xv.y * wr[j].y + xv.z * wr[j].z + xv.w * wr[j].w;
    }
    #pragma unroll
    for (int off = 16; off; off >>= 1) s += __shfl_xor(s, off, 32);
    if (lane == 0) {
      const bool mk = mask[(size_t)b * T + t0 + t] != 0;
      score_s[t] = mk ? (s + bb) : NEGV;
    }
  }
  __syncthreads();

  // ---- Phase 2: chunk-local softmax stats (wave 0) ------------------------
  if (wv == 0) {
    const float s = score_s[lane];
    float m = s;
    #pragma unroll
    for (int off = 16; off; off >>= 1) m = fmaxf(m, __shfl_xor(m, off, 32));
    const float p = (s > -1e8f) ? expf(s - m) : 0.f;  // fully-masked chunk -> all zeros
    alpha_s[lane] = p;
    float l = p;
    #pragma unroll
    for (int off = 16; off; off >>= 1) l += __shfl_xor(l, off, 32);
    if (lane == 0) {
      wsM[(size_t)b * C + c] = m;
      wsL[(size_t)b * C + c] = l;
    }
  }
  __syncthreads();

  float* accOut = wsAcc + ((size_t)b * C + c) * DMODEL;

  // ---- Phase 3: weighted accumulation partial[d] = sum_t alpha_t * x[t,d] -
#ifdef HAVE_WMMA4
  // Wave wv owns columns [128*wv, 128*wv+128): 8 N-tiles of 16, K in 8 steps of 4.
  // A(16x4 f32): lane<16 holds K={k0,k0+1}, lane>=16 holds K={k0+2,k0+3}; A[m,k]=alpha[k].
  // B(4x16 f32): lane holds N = lane&15 for the same K split.
  const int col0base = wv * 128;
  const int kb = (lane >= 16) ? 2 : 0;
  const int n  = lane & 15;
  v8f acc[8];
  #pragma unroll
  for (int tle = 0; tle < 8; ++tle) {
    #pragma unroll
    for (int e = 0; e < 8; ++e) acc[tle][e] = 0.f;
  }
  #pragma unroll
  for (int step = 0; step < 8; ++step) {
    const int k0 = step * 4;
    v2f a;
    a.x = alpha_s[k0 + kb];
    a.y = alpha_s[k0 + kb + 1];
    #pragma unroll
    for (int tle = 0; tle < 8; ++tle) {
      const int col0 = col0base + tle * 16;
      v2f bm;
      bm.x = xs[(k0 + kb)     * DMODEL + col0 + n];
      bm.y = xs[(k0 + kb + 1) * DMODEL + col0 + n];
      acc[tle] = __builtin_amdgcn_wmma_f32_16x16x4_f32(
          false, a, false, bm, (short)0, acc[tle], false, false);
    }
  }
  // D[m,n] identical over m (A broadcast); VGPR0, lanes 0..15 carry cols col0+n.
  if (lane < 16) {
    #pragma unroll
    for (int tle = 0; tle < 8; ++tle)
      accOut[col0base + tle * 16 + n] = acc[tle][0];
  }
#else
  // VALU fallback: each thread owns 4 contiguous columns.
  {
    const int col = tid * 4;
    float4 r = {0.f, 0.f, 0.f, 0.f};
    #pragma unroll
    for (int t = 0; t < TCH; ++t) {
      const float al = alpha_s[t];
      const float4 xv = *(const float4*)(xs + t * DMODEL + col);
      r.x += al * xv.x; r.y += al * xv.y; r.z += al * xv.z; r.w += al * xv.w;
    }
    *(float4*)(accOut + col) = r;
  }
#endif
}

// ---------------------------------------------------------------------------
// Kernel 2: one workgroup per batch row; merge chunk partials:
//   out[b,d] = sum_c e^{m_c-M} acc_c[d] / sum_c e^{m_c-M} l_c
// ---------------------------------------------------------------------------
__global__ __launch_bounds__(256) void app_combine_kernel(
    const float* __restrict__ wsAcc, const float* __restrict__ wsM,
    const float* __restrict__ wsL, float* __restrict__ out, int C)
{
  __shared__ float mv[MAXC], lv[MAXC], sc[MAXC];
  const int b = blockIdx.x, tid = threadIdx.x;
  if (tid < C) { mv[tid] = wsM[(size_t)b * C + tid]; lv[tid] = wsL[(size_t)b * C + tid]; }
  __syncthreads();
  float M = -3.0e38f;
  for (int i = 0; i < C; ++i) M = fmaxf(M, mv[i]);
  if (tid < C) sc[tid] = expf(mv[tid] - M);
  __syncthreads();
  float L = 0.f;
  for (int i = 0; i < C; ++i) L += sc[i] * lv[i];
  const float invL = 1.0f / L;   // t=0 always valid => L > 0
  for (int col = tid; col < DMODEL; col += 256) {
    float s = 0.f;
    for (int i = 0; i < C; ++i)
      s += sc[i] * wsAcc[((size_t)b * C + i) * DMODEL + col];
    out[(size_t)b * DMODEL + col] = s * invL;
  }
}

// ---------------------------------------------------------------------------
extern "C" void kernel_launch(void* const* d_in, const int* in_sizes, int n_in,
                              void* d_out, int out_size, void* d_ws, size_t ws_size,
                              hipStream_t stream) {
  (void)n_in; (void)ws_size;
  const float*         x    = (const float*)d_in[0];
  const unsigned char* mask = (const unsigned char*)d_in[1];  // jax bool = 1 byte
  const float*         w    = (const float*)d_in[2];
  const float*         bias = (const float*)d_in[3];
  float*               out  = (float*)d_out;

  const int d = in_sizes[2];            // 1024 (== DMODEL)
  const int B = out_size / d;           // 32
  const int T = in_sizes[1] / B;        // 2048
  const int C = T / TCH;                // 64 chunks per batch row

  // workspace layout (floats): acc[B*C*DMODEL] | m[B*C] | l[B*C]  (~8.4 MB)
  float* wsAcc = (float*)d_ws;
  float* wsM   = wsAcc + (size_t)B * C * DMODEL;
  float* wsL   = wsM + (size_t)B * C;

  dim3 g1(C, B);
  app_partial_kernel<<<g1, 256, 0, stream>>>(x, mask, w, bias, wsAcc, wsM, wsL, T, C);
  app_combine_kernel<<<B, 256, 0, stream>>>(wsAcc, wsM, wsL, out, C);
}